// CorrelationNetwork_25237227831644
// MI455X (gfx1250) — compile-verified
//
#include <hip/hip_runtime.h>

typedef float v2f __attribute__((ext_vector_type(2)));
typedef float v8f __attribute__((ext_vector_type(8)));

static constexpr int kB = 4;
static constexpr int kC = 32;
static constexpr int kH = 32;
static constexpr int kW = 1024;
static constexpr int kPatch = 21;          // 21x21 displacement grid
static constexpr int kCHstride = kH * kW;  // elements between channels
static constexpr int kS = 70;              // LDS row stride (dwords): even + conflict-free
static constexpr int kTrash = kPatch * kS; // per-lane trash slots for invalid scatter lanes

// Workgroup = 128 threads (4 waves). Grid = (W/64, H, B).
// Wave (p = parity, st = u-subtile) computes out[w = wbase+32*st+2m+p, all pj].
__global__ __launch_bounds__(128)
void corr441_wmma_f32(const float* __restrict__ x1,
                      const float* __restrict__ x2,
                      float* __restrict__ out) {
  __shared__ float tile[kPatch * kS + 32];  // 21 rows * 70 dwords + 32 trash

  const int wave = threadIdx.x >> 5;   // 0..3
  const int lane = threadIdx.x & 31;
  const int p    = wave & 1;           // w parity
  const int st   = wave >> 1;          // u-subtile (0 or 1)
  const int wbase = blockIdx.x * 64;
  const int h = blockIdx.y;
  const int b = blockIdx.z;

  const int nm = lane & 15;            // N (or M) index within the tile
  const int hi = lane >> 4;            // lane half selects K/M offset

  // w2 range over all (st, s, nm, p) is [wbase-20, wbase+107]; interior chunks
  // never need per-lane bounds masks (uniform branch, no exec juggling).
  const bool interior = (wbase >= 20) && (wbase + 107 < kW);

  // ---- Load A = x1 tile once (lives across the pi loop) --------------------
  // V_WMMA_F32_16X16X4_F32 A layout: pair t: .x -> K=4t+2*hi, .y -> K+1, M=nm
  v2f Areg[8];
  {
    const int w = wbase + 32 * st + 2 * nm + p;           // always in [0,1024)
    const float* a = x1 + ((size_t)b * kC * kH + h) * kW + w;
    #pragma unroll
    for (int t = 0; t < 8; ++t) {
      const int c0 = 4 * t + 2 * hi;
      Areg[t].x = a[(size_t)c0 * kCHstride];
      Areg[t].y = a[(size_t)(c0 + 1) * kCHstride];
    }
  }

  const int svals[3] = {-10, 6, 22};   // shift set covering lags -10..10

  for (int pi = 0; pi < kPatch; ++pi) {
    const int hh = h + 2 * (pi - 10);
    const size_t orow0 =
        (((size_t)(b * kPatch + pi) * kPatch) * kH + h) * kW + wbase + 2 * lane;

    if (hh < 0 || hh >= kH) {
      // Entire pi-plane is zero-padded: stream zeros, skip compute (uniform).
      #pragma unroll
      for (int k = 0; k < 6; ++k) {
        const int row = wave + 4 * k;                 // pj
        if (row < kPatch) {
          v2f z; z.x = 0.0f; z.y = 0.0f;
          __builtin_nontemporal_store(z, (v2f*)(out + orow0 + (size_t)row * kH * kW));
        }
      }
      continue;
    }

    const float* b2 = x2 + ((size_t)b * kC * kH + hh) * kW;

    // ---- Load all 3 B tiles, branch-free -----------------------------------
    v2f Breg[3][8];
    if (interior) {
      #pragma unroll
      for (int si = 0; si < 3; ++si) {
        const int w2 = wbase + 32 * st + 2 * svals[si] + 2 * nm + p;
        const float* bp = b2 + w2;
        #pragma unroll
        for (int t = 0; t < 8; ++t) {
          const int c0 = 4 * t + 2 * hi;
          Breg[si][t].x = bp[(size_t)c0 * kCHstride];
          Breg[si][t].y = bp[(size_t)(c0 + 1) * kCHstride];
        }
      }
    } else {
      #pragma unroll
      for (int si = 0; si < 3; ++si) {
        const int w2 = wbase + 32 * st + 2 * svals[si] + 2 * nm + p;
        const bool ok = (w2 >= 0) && (w2 < kW);
        const float* bp = b2 + (ok ? w2 : 0);         // clamped, always valid
        #pragma unroll
        for (int t = 0; t < 8; ++t) {
          const int c0 = 4 * t + 2 * hi;
          const float vx = bp[(size_t)c0 * kCHstride];
          const float vy = bp[(size_t)(c0 + 1) * kCHstride];
          Breg[si][t].x = ok ? vx : 0.0f;              // v_cndmask, no branches
          Breg[si][t].y = ok ? vy : 0.0f;
        }
      }
    }

    // ---- 3 band-GEMMs + unconditional LDS scatter --------------------------
    #pragma unroll
    for (int si = 0; si < 3; ++si) {
      v8f acc = {0.f, 0.f, 0.f, 0.f, 0.f, 0.f, 0.f, 0.f};
      #pragma unroll
      for (int t = 0; t < 8; ++t) {
        acc = __builtin_amdgcn_wmma_f32_16x16x4_f32(
            false, Areg[t], false, Breg[si][t], (short)0, acc, false, false);
      }
      const int s = svals[si];
      #pragma unroll
      for (int r = 0; r < 8; ++r) {
        const int m   = r + 8 * hi;                   // D row: VGPR r -> M=r, r+8
        const int lag = s + nm - m;
        const bool valid = (lag >= -10) && (lag <= 10);
        const int wl = 32 * st + 2 * m + p;
        const int addr = valid ? ((lag + 10) * kS + wl) : (kTrash + lane);
        tile[addr] = acc[r];                          // plain ds_store_b32
      }
    }

    __syncthreads();

    // ---- Coalesced write-out: one full 64-dword row per wave as float2 -----
    #pragma unroll
    for (int k = 0; k < 6; ++k) {
      const int row = wave + 4 * k;                   // pj
      if (row < kPatch) {
        v2f v = *(const v2f*)(&tile[row * kS + 2 * lane]);
        v.x *= (1.0f / 32.0f);
        v.y *= (1.0f / 32.0f);
        v.x = (v.x > 0.0f) ? v.x : 0.1f * v.x;
        v.y = (v.y > 0.0f) ? v.y : 0.1f * v.y;
        __builtin_nontemporal_store(v, (v2f*)(out + orow0 + (size_t)row * kH * kW));
      }
    }

    __syncthreads();
  }
}

extern "C" void kernel_launch(void* const* d_in, const int* in_sizes, int n_in,
                              void* d_out, int out_size, void* d_ws, size_t ws_size,
                              hipStream_t stream) {
  const float* x1 = (const float*)d_in[0];
  const float* x2 = (const float*)d_in[1];
  float* out = (float*)d_out;
  dim3 grid(kW / 64, kH, kB);   // (16, 32, 4)
  corr441_wmma_f32<<<grid, dim3(128), 0, stream>>>(x1, x2, out);
}